// RandLANetSmall_34943853920371
// MI455X (gfx1250) — compile-verified
//
#include <hip/hip_runtime.h>
#include <cstdint>

// ---------------------------------------------------------------------------
// RandLA-Net-small forward for MI455X (gfx1250, wave32).
// Pairwise sq-distances and all MLP GEMMs run on V_WMMA_F32_16X16X4_F32.
// ---------------------------------------------------------------------------

typedef float v2f __attribute__((ext_vector_type(2)));
typedef float v4f __attribute__((ext_vector_type(4)));
typedef float v8f __attribute__((ext_vector_type(8)));

__device__ __forceinline__ v8f wmma4(v2f a, v2f b, v8f c) {
  // D = A(16x4) * B(4x16) + C(16x16), f32, wave32.
  return __builtin_amdgcn_wmma_f32_16x16x4_f32(false, a, false, b, (short)0, c,
                                               false, false);
}

// ---------------------------------------------------------------------------
// GEMM  y[M x Cout] = x[M x K] @ W[K x Cout] + bias, optional BN-stat accum.
// M multiple of 128 (256 thr = 8 waves * 16 rows). K multiple of 4.
// NCT = number of 16-column tiles (compile-time). GUARD = partial last tile
// (loads clamped -> branch-free K-loop; only stores/stats are guarded).
// A layout: lane<16 -> K {k,k+1}, lane>=16 -> {k+2,k+3}, M = lane&15.
// B layout: VGPR0/1 = K rows (k / k+1, +2 for hi half), N = lane&15.
// C layout: VGPR v -> row v (+8 for hi half), col = lane&15.
// ---------------------------------------------------------------------------
template <int NCT, bool GUARD>
__global__ void gemm_stats_kernel(const float* __restrict__ x, int ldx, int Kdim,
                                  const float* __restrict__ W,
                                  const float* __restrict__ bias,
                                  float* __restrict__ y, int ldy,
                                  int Cout, float* __restrict__ stats) {
  const int wave = threadIdx.x >> 5;
  const int lane = threadIdx.x & 31;
  const int cl = lane & 15;
  const bool hi = lane >= 16;
  const int koff = hi ? 2 : 0;
  const int r0 = blockIdx.x * 128 + wave * 16;

  // Branch-free load pointers (clamped column when GUARD).
  const float* xp = x + (long)(r0 + cl) * ldx + koff;
  const float* wp0[NCT];
  const float* wp1[NCT];
#pragma unroll
  for (int ct = 0; ct < NCT; ++ct) {
    int col = ct * 16 + cl;
    if (GUARD && col >= Cout) col = Cout - 1;
    wp0[ct] = W + (long)(hi ? 2 : 0) * Cout + col;
    wp1[ct] = W + (long)(hi ? 3 : 1) * Cout + col;
  }

  v8f acc[NCT];
#pragma unroll
  for (int ct = 0; ct < NCT; ++ct) acc[ct] = {};

  const int wstep = 4 * Cout;
  for (int k = 0; k < Kdim; k += 4) {
    const v2f a = *(const v2f*)xp;  // one global_load_b64 per K-step
    xp += 4;
#pragma unroll
    for (int ct = 0; ct < NCT; ++ct) {
      v2f b;
      b.x = *wp0[ct];
      b.y = *wp1[ct];
      wp0[ct] += wstep;
      wp1[ct] += wstep;
      acc[ct] = wmma4(a, b, acc[ct]);
    }
  }

  // Epilogue: bias, store, BN stats.
#pragma unroll
  for (int ct = 0; ct < NCT; ++ct) {
    const int col = ct * 16 + cl;
    const bool colok = !GUARD || (col < Cout);
    const int colc = (GUARD && col >= Cout) ? Cout - 1 : col;
    const float bcol = bias[colc];
    float lsum = 0.f, lsq = 0.f;
#pragma unroll
    for (int v = 0; v < 8; ++v) {
      const float val = acc[ct][v] + bcol;
      const int row = r0 + v + (hi ? 8 : 0);
      if (colok) y[(long)row * ldy + col] = val;
      lsum += val;
      lsq += val * val;
    }
    if (stats) {
      lsum += __shfl_xor(lsum, 16, 32);  // merge lane halves (same col)
      lsq  += __shfl_xor(lsq, 16, 32);
      if (!hi && colok) {
        atomicAdd(&stats[col], lsum);  // GLOBAL_ATOMIC_ADD_F32
        atomicAdd(&stats[64 + col], lsq);
      }
    }
  }
}

// ---------------------------------------------------------------------------
// BatchNorm (per channel over all M rows) + ReLU, in place.
// stats slot: [0..63]=sum, [64..127]=sumsq.
// ---------------------------------------------------------------------------
__global__ void bn_relu_kernel(float* __restrict__ y,
                               const float* __restrict__ stats,
                               const float* __restrict__ gamma,
                               const float* __restrict__ beta,
                               long n, int C, float invM) {
  const long gid = (long)blockIdx.x * blockDim.x + threadIdx.x;
  if (gid >= n) return;
  const int c = (int)(gid % C);
  const float mu = stats[c] * invM;
  float var = stats[64 + c] * invM - mu * mu;
  var = var < 0.f ? 0.f : var;
  float v = (y[gid] - mu) * rsqrtf(var + 1e-5f) * gamma[c] + beta[c];
  y[gid] = v > 0.f ? v : 0.f;
}

// z = agg * sigmoid(t) + agg
__global__ void attn_combine_kernel(const float* __restrict__ a,
                                    const float* __restrict__ t,
                                    float* __restrict__ z, long n) {
  const long gid = (long)blockIdx.x * blockDim.x + threadIdx.x;
  if (gid >= n) return;
  const float s = 1.f / (1.f + expf(-t[gid]));
  z[gid] = a[gid] * s + a[gid];
}

// ---------------------------------------------------------------------------
// Self-kNN (k=8) via WMMA distance tiles.
// pts: [Npts x 4] (xyz0), pt2: [Npts] squared norms. grid.y = batch.
// Each wave owns 32 query rows; per 16-target tile: 2 WMMAs, then
// d' = |t|^2 - 2 q.t staged in a wave-private transposed LDS region
// ([col][row], 36-float row pitch -> 16B-aligned ds_store_b128, bank-clean),
// then each lane scans its own row keeping a sorted top-8 in registers
// (static indices only -> no scratch). Npts multiple of 256.
// ---------------------------------------------------------------------------
__global__ void knn_self_kernel(const float* __restrict__ pts,
                                const float* __restrict__ pt2,
                                int Npts, int* __restrict__ knn) {
  __shared__ float lds[8][16][36];  // [wave][col][row padded to 36]
  const int b = blockIdx.y;
  pts += (long)b * Npts * 4;
  pt2 += (long)b * Npts;
  knn += (long)b * Npts * 8;

  const int wave = threadIdx.x >> 5;
  const int lane = threadIdx.x & 31;
  const int cl = lane & 15;
  const bool hi = lane >= 16;
  const int koff = hi ? 2 : 0;
  const int r0 = blockIdx.x * 256 + wave * 32;

  // A fragments are loop-invariant (query rows); 8B-aligned v2f loads.
  const v2f aA = *(const v2f*)(pts + (long)(r0 + cl) * 4 + koff);
  const v2f aB = *(const v2f*)(pts + (long)(r0 + 16 + cl) * 4 + koff);

  float bd[8];
  int bi[8];
#pragma unroll
  for (int s = 0; s < 8; ++s) {
    bd[s] = 3.4e38f;
    bi[s] = r0 + lane;
  }

  float* const lp = &lds[wave][cl][0];
  const int rb = hi ? 8 : 0;

  for (int t0 = 0; t0 < Npts; t0 += 16) {
    const v2f bb = *(const v2f*)(pts + (long)(t0 + cl) * 4 + koff);
    v8f zero = {};
    const v8f dA = wmma4(aA, bb, zero);
    const v8f dB = wmma4(aB, bb, zero);
    const float t2c = pt2[t0 + cl];

    v4f d0, d1, d2, d3;
#pragma unroll
    for (int v = 0; v < 4; ++v) {
      d0[v] = t2c - 2.f * dA[v];
      d1[v] = t2c - 2.f * dA[4 + v];
      d2[v] = t2c - 2.f * dB[v];
      d3[v] = t2c - 2.f * dB[4 + v];
    }
    // Rows rb..rb+7 (tile A) and 16+rb..16+rb+7 (tile B): 4x ds_store_b128.
    *(v4f*)(lp + rb) = d0;
    *(v4f*)(lp + rb + 4) = d1;
    *(v4f*)(lp + 16 + rb) = d2;
    *(v4f*)(lp + 16 + rb + 4) = d3;

    // Wave-private LDS region; DS ops are in order within a wave.
#pragma unroll
    for (int c = 0; c < 16; ++c) {
      const float d = lds[wave][c][lane];
      if (d < bd[7]) {
        bd[7] = d;
        bi[7] = t0 + c;
#pragma unroll
        for (int s = 7; s > 0; --s) {
          if (bd[s] < bd[s - 1]) {
            float td = bd[s]; bd[s] = bd[s - 1]; bd[s - 1] = td;
            int ti = bi[s];  bi[s] = bi[s - 1]; bi[s - 1] = ti;
          }
        }
      }
    }
  }

  const int q = r0 + lane;
#pragma unroll
  for (int s = 0; s < 8; ++s) knn[(long)q * 8 + s] = bi[s];
}

// ---------------------------------------------------------------------------
// Cross nearest-neighbor (argmin). qpts/tpts are [N x 4] xyz0. grid.y = b.
// Target tiles staged as float4 (x,y,z,|t|^2): b128 loads both ways.
// Nq, Nt multiples of 256.
// ---------------------------------------------------------------------------
__global__ void nn_cross_kernel(const float* __restrict__ qpts, int Nq,
                                const float* __restrict__ tpts, int Nt,
                                int* __restrict__ out) {
  __shared__ v4f tile[256];
  const int b = blockIdx.y;
  qpts += (long)b * Nq * 4;
  tpts += (long)b * Nt * 4;
  out += (long)b * Nq;

  const int i = blockIdx.x * 256 + threadIdx.x;
  const v4f q = *(const v4f*)(qpts + (long)i * 4);
  float best = 3.4e38f;
  int bidx = 0;
  for (int t0 = 0; t0 < Nt; t0 += 256) {
    v4f t = *(const v4f*)(tpts + (long)(t0 + threadIdx.x) * 4);
    t.w = t.x * t.x + t.y * t.y + t.z * t.z;
    tile[threadIdx.x] = t;
    __syncthreads();
    for (int c = 0; c < 256; ++c) {
      const v4f tc = tile[c];  // broadcast ds_load_b128
      const float d = tc.w - 2.f * (q.x * tc.x + q.y * tc.y + q.z * tc.z);
      if (d < best) { best = d; bidx = t0 + c; }
    }
    __syncthreads();
  }
  out[i] = bidx;
}

// ---------------------------------------------------------------------------
// Prep: build x0pad [P x 8] = (xyz | 4 feats | 0), coordspad [P x 4] = xyz0,
// pt2 [P], and zero-pad fc_start.W (7x16 -> 8x16).
// ---------------------------------------------------------------------------
__global__ void prep_kernel(const float* __restrict__ coords,
                            const float* __restrict__ feats,
                            const float* __restrict__ fcW,
                            float* __restrict__ x0pad, float* __restrict__ cpad,
                            float* __restrict__ pt2, float* __restrict__ wpad,
                            int P) {
  const int i = blockIdx.x * 256 + threadIdx.x;
  if (i < 128) wpad[i] = (i >> 4) < 7 ? fcW[i] : 0.f;
  if (i >= P) return;
  const float x = coords[(long)i * 3 + 0];
  const float y = coords[(long)i * 3 + 1];
  const float z = coords[(long)i * 3 + 2];
  v4f c; c.x = x; c.y = y; c.z = z; c.w = 0.f;
  *(v4f*)(cpad + (long)i * 4) = c;
  pt2[i] = x * x + y * y + z * z;
  const v4f f = *(const v4f*)(feats + (long)i * 4);
  v4f lo; lo.x = x; lo.y = y; lo.z = z; lo.w = f.x;
  v4f hiv; hiv.x = f.y; hiv.y = f.z; hiv.z = f.w; hiv.w = 0.f;
  *(v4f*)(x0pad + (long)i * 8) = lo;
  *(v4f*)(x0pad + (long)i * 8 + 4) = hiv;
}

// Gather xyz0 rows by int64 index (same index list per batch) + norms.
__global__ void gather_coords_kernel(const float* __restrict__ src, int Nin,
                                     const long long* __restrict__ idx, int Nout,
                                     float* __restrict__ dst, float* __restrict__ dn) {
  const int b = blockIdx.y;
  const int i = blockIdx.x * 256 + threadIdx.x;
  if (i >= Nout) return;
  const long j = (long)idx[i];
  const v4f c = *(const v4f*)(src + ((long)b * Nin + j) * 4);
  *(v4f*)(dst + ((long)b * Nout + i) * 4) = c;
  dn[(long)b * Nout + i] = c.x * c.x + c.y * c.y + c.z * c.z;
}

// Gather feature rows (width C) by int64 index. grid.y = batch.
__global__ void gather_rows_kernel(const float* __restrict__ src, int Nin,
                                   const long long* __restrict__ idx, int Nout,
                                   int C, float* __restrict__ dst) {
  const int b = blockIdx.y;
  const long gid = (long)blockIdx.x * 256 + threadIdx.x;
  if (gid >= (long)Nout * C) return;
  const int i = (int)(gid / C), c = (int)(gid % C);
  const long j = (long)idx[i];
  dst[((long)b * Nout + i) * C + c] = src[((long)b * Nin + j) * C + c];
}

// Mean over 8 kNN neighbors. knn holds batch-local indices. grid.y = batch.
__global__ void nbr_mean_kernel(const float* __restrict__ f,
                                const int* __restrict__ knn, int N, int C,
                                float* __restrict__ out) {
  const int b = blockIdx.y;
  const long gid = (long)blockIdx.x * 256 + threadIdx.x;
  if (gid >= (long)N * C) return;
  const int i = (int)(gid / C), c = (int)(gid % C);
  const int* kr = knn + ((long)b * N + i) * 8;
  float s = 0.f;
#pragma unroll
  for (int u = 0; u < 8; ++u) s += f[((long)b * N + kr[u]) * C + c];
  out[((long)b * N + i) * C + c] = s * 0.125f;
}

// dst[i] = concat(flow[i] (C1), fhigh[nn[i]] (C2)). grid.y = batch.
__global__ void concat_up_kernel(const float* __restrict__ flow, int C1,
                                 const float* __restrict__ fhigh, int C2,
                                 const int* __restrict__ nn, int Nq, int Nt,
                                 float* __restrict__ dst) {
  const int b = blockIdx.y;
  const int C = C1 + C2;
  const long gid = (long)blockIdx.x * 256 + threadIdx.x;
  if (gid >= (long)Nq * C) return;
  const int i = (int)(gid / C), c = (int)(gid % C);
  float v;
  if (c < C1) {
    v = flow[((long)b * Nq + i) * C1 + c];
  } else {
    const int j = nn[(long)b * Nq + i];
    v = fhigh[((long)b * Nt + j) * C2 + (c - C1)];
  }
  dst[((long)b * Nq + i) * C + c] = v;
}

// ---------------------------------------------------------------------------
// Host-side orchestration.
// ---------------------------------------------------------------------------
extern "C" void kernel_launch(void* const* d_in, const int* in_sizes, int n_in,
                              void* d_out, int out_size, void* d_ws, size_t ws_size,
                              hipStream_t stream) {
  (void)in_sizes; (void)n_in; (void)out_size; (void)ws_size;
  constexpr int B = 2, N = 8192;
  constexpr int P1 = B * N;                // 16384
  constexpr int N2 = N / 4, P2 = B * N2;   // 2048 / 4096
  constexpr int N3 = N / 16, P3 = B * N3;  // 512  / 1024

  // Input indices (JAX pytree dict flatten: sorted keys; per-MLP W,b,beta,gamma).
  const float* coords = (const float*)d_in[0];
  const float* feats  = (const float*)d_in[1];
  const long long* idx1 = (const long long*)d_in[2];  // int64 in reference
  const long long* idx2 = (const long long*)d_in[3];
  const float* cls_W = (const float*)d_in[4];
  const float* cls_b = (const float*)d_in[5];
  enum { CLS_MLP = 6, DEC1 = 10, DEC2 = 14, E1_ATTN = 18, E1_M1 = 22,
         E1_M2 = 26, E2_ATTN = 30, E2_M1 = 34, E2_M2 = 38, E3_ATTN = 42,
         E3_M1 = 46, E3_M2 = 50, FC_START = 54 };
  auto pW = [&](int base) { return (const float*)d_in[base + 0]; };
  auto pB = [&](int base) { return (const float*)d_in[base + 1]; };
  auto pBeta  = [&](int base) { return (const float*)d_in[base + 2]; };
  auto pGamma = [&](int base) { return (const float*)d_in[base + 3]; };

  // Workspace bump allocator (256-B aligned).
  size_t off = 0;
  auto alloc = [&](size_t nbytes) {
    off = (off + 255) & ~(size_t)255;
    void* p = (char*)d_ws + off;
    off += nbytes;
    return p;
  };
  float* stats  = (float*)alloc(13 * 128 * sizeof(float));
  float* x0pad  = (float*)alloc((size_t)P1 * 8 * sizeof(float));
  float* cpad   = (float*)alloc((size_t)P1 * 4 * sizeof(float));
  float* pt2_1  = (float*)alloc((size_t)P1 * sizeof(float));
  float* wpad   = (float*)alloc(128 * sizeof(float));
  float* f0     = (float*)alloc((size_t)P1 * 16 * sizeof(float));
  float* mean1  = (float*)alloc((size_t)P1 * 16 * sizeof(float));
  float* agg1   = (float*)alloc((size_t)P1 * 16 * sizeof(float));
  float* t1     = (float*)alloc((size_t)P1 * 16 * sizeof(float));
  float* z1     = (float*)alloc((size_t)P1 * 16 * sizeof(float));
  float* f1     = (float*)alloc((size_t)P1 * 32 * sizeof(float));
  float* c2     = (float*)alloc((size_t)P2 * 4 * sizeof(float));
  float* c2n    = (float*)alloc((size_t)P2 * sizeof(float));
  float* f1ds   = (float*)alloc((size_t)P2 * 32 * sizeof(float));
  float* mean2  = (float*)alloc((size_t)P2 * 32 * sizeof(float));
  float* agg2   = (float*)alloc((size_t)P2 * 32 * sizeof(float));
  float* t2     = (float*)alloc((size_t)P2 * 32 * sizeof(float));
  float* z2     = (float*)alloc((size_t)P2 * 32 * sizeof(float));
  float* f2     = (float*)alloc((size_t)P2 * 64 * sizeof(float));
  float* c3     = (float*)alloc((size_t)P3 * 4 * sizeof(float));
  float* c3n    = (float*)alloc((size_t)P3 * sizeof(float));
  float* f2ds   = (float*)alloc((size_t)P3 * 64 * sizeof(float));
  float* mean3  = (float*)alloc((size_t)P3 * 64 * sizeof(float));
  float* agg3   = (float*)alloc((size_t)P3 * 32 * sizeof(float));
  float* t3     = (float*)alloc((size_t)P3 * 32 * sizeof(float));
  float* z3     = (float*)alloc((size_t)P3 * 32 * sizeof(float));
  float* f3     = (float*)alloc((size_t)P3 * 64 * sizeof(float));
  float* dcat2  = (float*)alloc((size_t)P2 * 128 * sizeof(float));
  float* fdec2  = (float*)alloc((size_t)P2 * 32 * sizeof(float));
  float* dcat1  = (float*)alloc((size_t)P1 * 64 * sizeof(float));
  float* fdec1  = (float*)alloc((size_t)P1 * 16 * sizeof(float));
  float* hbuf   = (float*)alloc((size_t)P1 * 16 * sizeof(float));
  int* knn1 = (int*)alloc((size_t)P1 * 8 * sizeof(int));
  int* knn2 = (int*)alloc((size_t)P2 * 8 * sizeof(int));
  int* knn3 = (int*)alloc((size_t)P3 * 8 * sizeof(int));
  int* nn23 = (int*)alloc((size_t)P2 * sizeof(int));
  int* nn12 = (int*)alloc((size_t)P1 * sizeof(int));

  hipMemsetAsync(stats, 0, 13 * 128 * sizeof(float), stream);

  auto launch_gemm = [&](const float* x, int ldx, int K, const float* W,
                         const float* bias, float* y, int ldy, int Cout,
                         float* st, int M) {
    const dim3 g(M / 128);
    if (Cout == 16)
      gemm_stats_kernel<1, false><<<g, 256, 0, stream>>>(x, ldx, K, W, bias, y, ldy, Cout, st);
    else if (Cout == 32)
      gemm_stats_kernel<2, false><<<g, 256, 0, stream>>>(x, ldx, K, W, bias, y, ldy, Cout, st);
    else if (Cout == 64)
      gemm_stats_kernel<4, false><<<g, 256, 0, stream>>>(x, ldx, K, W, bias, y, ldy, Cout, st);
    else
      gemm_stats_kernel<1, true><<<g, 256, 0, stream>>>(x, ldx, K, W, bias, y, ldy, Cout, st);
  };

  int slot = 0;
  auto mlp = [&](const float* x, int ldx, int K, int base, float* y, int M, int C) {
    float* st = stats + (slot++) * 128;
    launch_gemm(x, ldx, K, pW(base), pB(base), y, C, C, st, M);
    const long n = (long)M * C;
    bn_relu_kernel<<<dim3((unsigned)((n + 255) / 256)), 256, 0, stream>>>(
        y, st, pGamma(base), pBeta(base), n, C, 1.f / (float)M);
  };
  auto ew = [&](long n) { return dim3((unsigned)((n + 255) / 256)); };

  // ---- Stage 0: prep + fc_start --------------------------------------------
  prep_kernel<<<ew(P1), 256, 0, stream>>>(coords, feats, pW(FC_START), x0pad,
                                          cpad, pt2_1, wpad, P1);
  {
    float* st = stats + (slot++) * 128;
    launch_gemm(x0pad, 8, 8, wpad, pB(FC_START), f0, 16, 16, st, P1);
    bn_relu_kernel<<<ew((long)P1 * 16), 256, 0, stream>>>(
        f0, st, pGamma(FC_START), pBeta(FC_START), (long)P1 * 16, 16, 1.f / P1);
  }

  // ---- Level 1 LFA ---------------------------------------------------------
  knn_self_kernel<<<dim3(N / 256, B), 256, 0, stream>>>(cpad, pt2_1, N, knn1);
  nbr_mean_kernel<<<dim3((unsigned)(((long)N * 16 + 255) / 256), B), 256, 0, stream>>>(
      f0, knn1, N, 16, mean1);
  mlp(mean1, 16, 16, E1_M1, agg1, P1, 16);
  mlp(agg1, 16, 16, E1_ATTN, t1, P1, 16);
  attn_combine_kernel<<<ew((long)P1 * 16), 256, 0, stream>>>(agg1, t1, z1, (long)P1 * 16);
  mlp(z1, 16, 16, E1_M2, f1, P1, 32);

  // ---- Downsample 1, Level 2 LFA ------------------------------------------
  gather_coords_kernel<<<dim3(N2 / 256, B), 256, 0, stream>>>(cpad, N, idx1, N2, c2, c2n);
  gather_rows_kernel<<<dim3((unsigned)(((long)N2 * 32 + 255) / 256), B), 256, 0, stream>>>(
      f1, N, idx1, N2, 32, f1ds);
  knn_self_kernel<<<dim3(N2 / 256, B), 256, 0, stream>>>(c2, c2n, N2, knn2);
  nbr_mean_kernel<<<dim3((unsigned)(((long)N2 * 32 + 255) / 256), B), 256, 0, stream>>>(
      f1ds, knn2, N2, 32, mean2);
  mlp(mean2, 32, 32, E2_M1, agg2, P2, 32);
  mlp(agg2, 32, 32, E2_ATTN, t2, P2, 32);
  attn_combine_kernel<<<ew((long)P2 * 32), 256, 0, stream>>>(agg2, t2, z2, (long)P2 * 32);
  mlp(z2, 32, 32, E2_M2, f2, P2, 64);

  // ---- Downsample 2, Level 3 LFA ------------------------------------------
  gather_coords_kernel<<<dim3(1, B), 256, 0, stream>>>(c2, N2, idx2, N3, c3, c3n);
  gather_rows_kernel<<<dim3((unsigned)(((long)N3 * 64 + 255) / 256), B), 256, 0, stream>>>(
      f2, N2, idx2, N3, 64, f2ds);
  knn_self_kernel<<<dim3(N3 / 256, B), 256, 0, stream>>>(c3, c3n, N3, knn3);
  nbr_mean_kernel<<<dim3((unsigned)(((long)N3 * 64 + 255) / 256), B), 256, 0, stream>>>(
      f2ds, knn3, N3, 64, mean3);
  mlp(mean3, 64, 64, E3_M1, agg3, P3, 32);
  mlp(agg3, 32, 32, E3_ATTN, t3, P3, 32);
  attn_combine_kernel<<<ew((long)P3 * 32), 256, 0, stream>>>(agg3, t3, z3, (long)P3 * 32);
  mlp(z3, 32, 32, E3_M2, f3, P3, 64);

  // ---- Decoder -------------------------------------------------------------
  nn_cross_kernel<<<dim3(N2 / 256, B), 256, 0, stream>>>(c2, N2, c3, N3, nn23);
  concat_up_kernel<<<dim3((unsigned)(((long)N2 * 128 + 255) / 256), B), 256, 0, stream>>>(
      f2, 64, f3, 64, nn23, N2, N3, dcat2);
  mlp(dcat2, 128, 128, DEC2, fdec2, P2, 32);

  nn_cross_kernel<<<dim3(N / 256, B), 256, 0, stream>>>(cpad, N, c2, N2, nn12);
  concat_up_kernel<<<dim3((unsigned)(((long)N * 64 + 255) / 256), B), 256, 0, stream>>>(
      f1, 32, fdec2, 32, nn12, N, N2, dcat1);
  mlp(dcat1, 64, 64, DEC1, fdec1, P1, 16);
  mlp(fdec1, 16, 16, CLS_MLP, hbuf, P1, 16);

  // ---- Classifier: out = h @ cls_W + cls_b  (16 -> 9, no BN) ---------------
  launch_gemm(hbuf, 16, 16, cls_W, cls_b, (float*)d_out, 9, 9, nullptr, P1);
}